// GAT_46076409152403
// MI455X (gfx1250) — compile-verified
//
#include <hip/hip_runtime.h>

#define NODES   50000
#define EDGES   400000
#define EE      (EDGES + NODES)   // +self loops = 450000
#define DIN     256
#define DH      32
#define HEADS   8
#define C1      (HEADS * DH)      // 256
#define DOUT    40
#define SLOPE   0.2f

typedef __attribute__((ext_vector_type(2))) float v2f;
typedef __attribute__((ext_vector_type(8))) float v8f;

__device__ __forceinline__ unsigned fkey(float v) {
  unsigned u = __float_as_uint(v);
  return (u & 0x80000000u) ? ~u : (u | 0x80000000u);
}
__device__ __forceinline__ float unkey(unsigned k) {
  return (k & 0x80000000u) ? __uint_as_float(k & 0x7fffffffu)
                           : __uint_as_float(~k);
}
__device__ __forceinline__ float lrelu(float v) { return v > 0.f ? v : SLOPE * v; }

// ---------------------------------------------------------------- zero
__global__ __launch_bounds__(256)
void k_zero(float* __restrict__ p, int n) {
  int i = blockIdx.x * blockDim.x + threadIdx.x;
  if (i < n) p[i] = 0.f;
}

// ---------------------------------------------------------------- fp32 WMMA GEMM
// C[M,NN] = A[M,K] @ W[K,NN] + bias[NN].  One wave -> NT adjacent 16x16 N-tiles
// of one 16-row M stripe; the A fragment is loaded once per k-step and reused
// for NT independent WMMAs.  All B loads of a k-step are issued before any
// WMMA consumes them (distinct registers -> clause + single wait).
// Out-of-range N columns: address clamped, accumulator garbage discarded at
// the guarded store (D column n depends only on B column n).
template <int NT, int NN>
__global__ __launch_bounds__(256)
void k_gemm_wmma(const float* __restrict__ A, const float* __restrict__ W,
                 const float* __restrict__ bias, float* __restrict__ Cout,
                 int K, int groupsN, int totalGroups) {
  int grp = (blockIdx.x << 3) + (threadIdx.x >> 5);
  if (grp >= totalGroups) return;            // whole-wave exit, EXEC stays full
  int tm = grp / groupsN;
  int tg = grp - tm * groupsN;               // which group of NT tiles along N
  int lane = threadIdx.x & 31;
  int l15  = lane & 15;
  int kh   = (lane >> 4) << 1;               // 0 or 2 (K sub-phase per half-wave)
  int row  = tm * 16 + l15;
  const float* arow = A + (size_t)row * K + kh;

  int col[NT];
  bool ok[NT];
  const float* pb[NT];
  #pragma unroll
  for (int t = 0; t < NT; ++t) {
    col[t] = (tg * NT + t) * 16 + l15;
    ok[t]  = col[t] < NN;
    pb[t]  = W + (size_t)kh * NN + (ok[t] ? col[t] : 0);   // clamped, never OOB
  }

  v8f zero8 = {0.f, 0.f, 0.f, 0.f, 0.f, 0.f, 0.f, 0.f};
  v8f acc[NT];
  #pragma unroll
  for (int t = 0; t < NT; ++t) acc[t] = zero8;

  #pragma unroll 2
  for (int k0 = 0; k0 < K; k0 += 4) {
    v2f a = *(const v2f*)(arow + k0);        // A[row][k0+kh], A[row][k0+kh+1]
    float bx[NT], by[NT];
    #pragma unroll
    for (int t = 0; t < NT; ++t) {           // issue all B loads first
      bx[t] = pb[t][0];                      // W[k0+kh  ][col]  (imm offset 0)
      by[t] = pb[t][NN];                     // W[k0+kh+1][col]  (imm offset NN)
    }
    #pragma unroll
    for (int t = 0; t < NT; ++t) {
      v2f b;
      b.x = bx[t];
      b.y = by[t];
      acc[t] = __builtin_amdgcn_wmma_f32_16x16x4_f32(
          false, a, false, b, (short)0, acc[t], false, false);
      pb[t] += 4 * NN;                       // constant stride advance
    }
  }

  int rbase = tm * 16 + ((lane >> 4) << 3);  // rows r / r+8 per half-wave
  #pragma unroll
  for (int t = 0; t < NT; ++t) {
    if (ok[t]) {
      float bv = bias[col[t]];
      #pragma unroll
      for (int r = 0; r < 8; ++r)
        Cout[(size_t)(rbase + r) * NN + col[t]] = acc[t][r] + bv;
    }
  }
}

// ---------------------------------------------------------------- layer-1 edge logits
// wave = one edge; 256 channels via 2 iters of float4/lane; 8-lane groups = one head
__global__ __launch_bounds__(256)
void k_edge_logits_h8(const float* __restrict__ xl, const float* __restrict__ xr,
                      const float* __restrict__ att, const int* __restrict__ ei,
                      float* __restrict__ sout, unsigned* __restrict__ nmaxkey) {
  int wid  = (blockIdx.x * blockDim.x + threadIdx.x) >> 5;
  int lane = threadIdx.x & 31;
  if (wid >= EE) return;
  int s, d;
  if (wid < EDGES) { s = ei[wid]; d = ei[EDGES + wid]; }
  else             { s = wid - EDGES; d = s; }
  const float* pl = xl + (size_t)s * C1;
  const float* pr = xr + (size_t)d * C1;
  #pragma unroll
  for (int iter = 0; iter < 2; ++iter) {
    int c0 = iter * 128 + lane * 4;
    float4 a = *(const float4*)(pl + c0);
    float4 b = *(const float4*)(pr + c0);
    float4 w = *(const float4*)(att + c0);
    float v = lrelu(a.x + b.x) * w.x + lrelu(a.y + b.y) * w.y
            + lrelu(a.z + b.z) * w.z + lrelu(a.w + b.w) * w.w;
    v += __shfl_xor(v, 1);
    v += __shfl_xor(v, 2);
    v += __shfl_xor(v, 4);                   // sum within 8-lane head group
    if ((lane & 7) == 0) {
      int h = iter * 4 + (lane >> 3);
      sout[(size_t)wid * HEADS + h] = v;
      atomicMax(&nmaxkey[(size_t)d * HEADS + h], fkey(v));
    }
  }
}

// ---------------------------------------------------------------- exp + denom
template <int H>
__global__ __launch_bounds__(256)
void k_edge_exp(float* __restrict__ sbuf, const unsigned* __restrict__ nmaxkey,
                float* __restrict__ denom, const int* __restrict__ ei) {
  int tid = blockIdx.x * blockDim.x + threadIdx.x;
  if (tid >= EE * H) return;
  int e = tid / H, h = tid - e * H;
  int d = (e < EDGES) ? ei[EDGES + e] : (e - EDGES);
  float m  = unkey(nmaxkey[(size_t)d * H + h]);
  float ex = expf(sbuf[tid] - m);
  sbuf[tid] = ex;
  atomicAdd(&denom[(size_t)d * H + h], ex);
}

// ---------------------------------------------------------------- weighted scatter
__global__ __launch_bounds__(256)
void k_scatter1(const float* __restrict__ xl, const float* __restrict__ ex,
                const float* __restrict__ denom, const int* __restrict__ ei,
                float* __restrict__ acc) {
  long tid = (long)blockIdx.x * blockDim.x + threadIdx.x;
  if (tid >= (long)EE * C1) return;
  int e = (int)(tid >> 8);
  int c = (int)(tid & 255);
  int s, d;
  if (e < EDGES) { s = ei[e]; d = ei[EDGES + e]; }
  else           { s = e - EDGES; d = s; }
  int h = c >> 5;
  float alpha = ex[(size_t)e * HEADS + h] / denom[(size_t)d * HEADS + h];
  atomicAdd(&acc[(size_t)d * C1 + c], alpha * xl[(size_t)s * C1 + c]);
}

__global__ __launch_bounds__(256)
void k_scatter2(const float* __restrict__ hl, const float* __restrict__ ex,
                const float* __restrict__ denom, const int* __restrict__ ei,
                float* __restrict__ acc) {
  long tid = (long)blockIdx.x * blockDim.x + threadIdx.x;
  if (tid >= (long)EE * DOUT) return;
  int e = (int)(tid / DOUT);
  int c = (int)(tid - (long)e * DOUT);
  int s, d;
  if (e < EDGES) { s = ei[e]; d = ei[EDGES + e]; }
  else           { s = e - EDGES; d = s; }
  float alpha = ex[e] / denom[d];
  atomicAdd(&acc[(size_t)d * DOUT + c], alpha * hl[(size_t)s * DOUT + c]);
}

// ---------------------------------------------------------------- ELU + bias (in place)
__global__ __launch_bounds__(256)
void k_elu_bias(float* __restrict__ acc, const float* __restrict__ bias, int n) {
  int i = blockIdx.x * blockDim.x + threadIdx.x;
  if (i >= n) return;
  float v = acc[i] + bias[i & (C1 - 1)];
  acc[i] = v > 0.f ? v : expm1f(v);
}

// ---------------------------------------------------------------- layer-2 edge logits (1 head)
__global__ __launch_bounds__(256)
void k_edge_logits_h1(const float* __restrict__ hl, const float* __restrict__ hr,
                      const float* __restrict__ att, const int* __restrict__ ei,
                      float* __restrict__ sout, unsigned* __restrict__ nmaxkey) {
  int wid  = (blockIdx.x * blockDim.x + threadIdx.x) >> 5;
  int lane = threadIdx.x & 31;
  if (wid >= EE) return;
  int s, d;
  if (wid < EDGES) { s = ei[wid]; d = ei[EDGES + wid]; }
  else             { s = wid - EDGES; d = s; }
  const float* pl = hl + (size_t)s * DOUT;
  const float* pr = hr + (size_t)d * DOUT;
  float v = lrelu(pl[lane] + pr[lane]) * att[lane];
  if (lane < DOUT - 32)
    v += lrelu(pl[32 + lane] + pr[32 + lane]) * att[32 + lane];
  v += __shfl_xor(v, 1);
  v += __shfl_xor(v, 2);
  v += __shfl_xor(v, 4);
  v += __shfl_xor(v, 8);
  v += __shfl_xor(v, 16);
  if (lane == 0) {
    sout[wid] = v;
    atomicMax(&nmaxkey[d], fkey(v));
  }
}

// ---------------------------------------------------------------- log_softmax(40) per node
__global__ __launch_bounds__(256)
void k_logsoftmax(const float* __restrict__ acc, const float* __restrict__ bias,
                  float* __restrict__ out) {
  int wid  = (blockIdx.x * blockDim.x + threadIdx.x) >> 5;
  int lane = threadIdx.x & 31;
  if (wid >= NODES) return;
  const float* p = acc + (size_t)wid * DOUT;
  bool has2 = lane < DOUT - 32;
  float v0 = p[lane] + bias[lane];
  float v1 = has2 ? (p[32 + lane] + bias[32 + lane]) : -__builtin_inff();
  float m = fmaxf(v0, v1);
  for (int o = 1; o < 32; o <<= 1) m = fmaxf(m, __shfl_xor(m, o));
  float se = expf(v0 - m) + (has2 ? expf(v1 - m) : 0.f);
  for (int o = 1; o < 32; o <<= 1) se += __shfl_xor(se, o);
  float lse = m + logf(se);
  out[(size_t)wid * DOUT + lane] = v0 - lse;
  if (has2) out[(size_t)wid * DOUT + 32 + lane] = v1 - lse;
}

// ================================================================ launch
extern "C" void kernel_launch(void* const* d_in, const int* in_sizes, int n_in,
                              void* d_out, int out_size, void* d_ws, size_t ws_size,
                              hipStream_t stream) {
  const float* x     = (const float*)d_in[0];
  const int*   ei    = (const int*)d_in[1];
  const float* W1l   = (const float*)d_in[2];
  const float* b1l   = (const float*)d_in[3];
  const float* W1r   = (const float*)d_in[4];
  const float* b1r   = (const float*)d_in[5];
  const float* att1  = (const float*)d_in[6];
  const float* bias1 = (const float*)d_in[7];
  const float* W2l   = (const float*)d_in[8];
  const float* b2l   = (const float*)d_in[9];
  const float* W2r   = (const float*)d_in[10];
  const float* b2r   = (const float*)d_in[11];
  const float* att2  = (const float*)d_in[12];
  const float* bias2 = (const float*)d_in[13];
  float* out = (float*)d_out;
  float* ws  = (float*)d_ws;

  const size_t NM = (size_t)NODES * C1;             // 12.8M floats
  float*    xl1   = ws;
  float*    xr1   = ws + NM;
  float*    acc1  = ws + 2 * NM;                    // later reused as h (in-place ELU)
  unsigned* nmax1 = (unsigned*)(ws + 3 * NM);
  float*    den1  = ws + 3 * NM + (size_t)NODES * HEADS;
  float*    s1    = ws + 3 * NM + 2 * (size_t)NODES * HEADS;  // EE*8 floats
  // layer-2 overlays on dead layer-1 buffers
  float*    hl2   = xl1;
  float*    hr2   = xr1;
  float*    acc2  = s1;                                           // NODES*40
  unsigned* nmax2 = (unsigned*)(s1 + (size_t)NODES * DOUT);       // NODES
  float*    den2  = s1 + (size_t)NODES * DOUT + NODES;            // NODES
  float*    s2    = s1 + (size_t)NODES * DOUT + 2 * (size_t)NODES;// EE

  // ---- layer 1 GEMMs (WMMA f32, 4 N-tiles per wave) ----
  {
    int groupsN = (C1 / 16) / 4;                       // 4
    int totalGroups = (NODES / 16) * groupsN;          // 12500
    int blocks = (totalGroups + 7) / 8;
    k_gemm_wmma<4, C1><<<blocks, 256, 0, stream>>>(x, W1l, b1l, xl1, DIN, groupsN, totalGroups);
    k_gemm_wmma<4, C1><<<blocks, 256, 0, stream>>>(x, W1r, b1r, xr1, DIN, groupsN, totalGroups);
  }
  // zero acc1 | nmax1 | den1 (contiguous)
  {
    int n = (int)(NM + 2 * (size_t)NODES * HEADS);
    k_zero<<<(n + 255) / 256, 256, 0, stream>>>(acc1, n);
  }
  // ---- layer 1 edge softmax + aggregate ----
  {
    long t = (long)EE * 32;
    k_edge_logits_h8<<<(int)((t + 255) / 256), 256, 0, stream>>>(xl1, xr1, att1, ei, s1, nmax1);
    int n = EE * HEADS;
    k_edge_exp<HEADS><<<(n + 255) / 256, 256, 0, stream>>>(s1, nmax1, den1, ei);
    long ns = (long)EE * C1;
    k_scatter1<<<(int)((ns + 255) / 256), 256, 0, stream>>>(xl1, s1, den1, ei, acc1);
  }
  k_elu_bias<<<(int)((NM + 255) / 256), 256, 0, stream>>>(acc1, bias1, (int)NM);

  // ---- layer 2 GEMMs (N=40 -> 3 guarded tiles per wave) ----
  {
    int groupsN = 1;
    int totalGroups = (NODES / 16);                    // 3125
    int blocks = (totalGroups + 7) / 8;
    k_gemm_wmma<3, DOUT><<<blocks, 256, 0, stream>>>(acc1, W2l, b2l, hl2, C1, groupsN, totalGroups);
    k_gemm_wmma<3, DOUT><<<blocks, 256, 0, stream>>>(acc1, W2r, b2r, hr2, C1, groupsN, totalGroups);
  }
  // zero acc2 | nmax2 | den2 (contiguous)
  {
    int n = NODES * DOUT + 2 * NODES;
    k_zero<<<(n + 255) / 256, 256, 0, stream>>>(acc2, n);
  }
  // ---- layer 2 edge softmax + aggregate ----
  {
    long t = (long)EE * 32;
    k_edge_logits_h1<<<(int)((t + 255) / 256), 256, 0, stream>>>(hl2, hr2, att2, ei, s2, nmax2);
    k_edge_exp<1><<<(EE + 255) / 256, 256, 0, stream>>>(s2, nmax2, den2, ei);
    long ns = (long)EE * DOUT;
    k_scatter2<<<(int)((ns + 255) / 256), 256, 0, stream>>>(hl2, s2, den2, ei, acc2);
  }
  // ---- final log_softmax ----
  {
    long t = (long)NODES * 32;
    k_logsoftmax<<<(int)((t + 255) / 256), 256, 0, stream>>>(acc2, bias2, out);
  }
}